// LearnableSparseTrigger_16286515987242
// MI455X (gfx1250) — compile-verified
//
#include <hip/hip_runtime.h>
#include <cstdint>
#include <cstddef>

#define L_TOTAL   4096
#define SEG_LENC  1024
#define NUM_SEGC  4
#define HALF_WIN  512
#define NVALID    3073      /* L - SEG + 1 */
#define MAX_START 3072
#define GSHIFT    16

typedef float v2f __attribute__((ext_vector_type(2)));
typedef float v8f __attribute__((ext_vector_type(8)));

// ---------------- gfx1250 async global->LDS helpers (ASYNCcnt-tracked) ------
__device__ __forceinline__ void async_g2l_b128(uint32_t lds_byte,
                                               const void* sbase,
                                               uint32_t voff_bytes) {
#if defined(__gfx1250__)
  unsigned long long sb = (unsigned long long)(uintptr_t)sbase;
  asm volatile("global_load_async_to_lds_b128 %0, %1, %2"
               :: "v"(lds_byte), "v"(voff_bytes), "s"(sb)
               : "memory");
#else
  (void)lds_byte; (void)sbase; (void)voff_bytes;
#endif
}

__device__ __forceinline__ void wait_async0() {
#if defined(__gfx1250__)
#if __has_builtin(__builtin_amdgcn_s_wait_asynccnt)
  __builtin_amdgcn_s_wait_asynccnt(0);
#else
  asm volatile("s_wait_asynccnt 0x0" ::: "memory");
#endif
#endif
}

// ---------------- Kernel P: tanh(pattern), relu(scales) into workspace ------
__global__ __launch_bounds__(256)
void lst_pattern_kernel(const float* __restrict__ pat_i,
                        const float* __restrict__ pat_q,
                        const float* __restrict__ seg_scale,
                        float* __restrict__ ws_pat,     // 2048 floats
                        float* __restrict__ ws_scales)  // 4 floats
{
  int i = blockIdx.x * 256 + threadIdx.x;
  if (i < SEG_LENC)            ws_pat[i] = tanhf(pat_i[i]);
  else if (i < 2 * SEG_LENC)   ws_pat[i] = tanhf(pat_q[i - SEG_LENC]);
  if (i < NUM_SEGC)            ws_scales[i] = fmaxf(seg_scale[i], 0.0f);
}

// ---------------- Kernel A: power scan + window means + sort + starts -------
__global__ __launch_bounds__(256)
void lst_select_kernel(const float* __restrict__ x,
                       const int*   __restrict__ rand_cols,
                       const int*   __restrict__ shifts_raw,
                       int*   __restrict__ ws_starts,   // [B*4]
                       float* __restrict__ ws_amp)      // [B]
{
  __shared__ __align__(16) float sA[2 * L_TOTAL]; // 32 KB: x stage -> sort keys
  __shared__ __align__(16) float sS[L_TOTAL];     // 16 KB: exclusive prefix of power
  __shared__ float sPart[256];
  __shared__ float sRP[256];
  __shared__ float sMisc[4];

  const int tid = threadIdx.x;
  const int b   = blockIdx.x;
  const float* xrow = x + (size_t)b * (2 * L_TOTAL);

  // Stage x[b,0,:] and x[b,1,:] (32 KB contiguous) into LDS via async copies.
  {
    uint32_t lds0 = (uint32_t)(uintptr_t)&sA[0];
#pragma unroll
    for (int c = 0; c < 8; ++c) {
      uint32_t off = (uint32_t)(c * 256 + tid) * 16u;
      async_g2l_b128(lds0 + off, xrow, off);
    }
    wait_async0();
  }
  __syncthreads();

  // Per-thread power over 16 consecutive samples + local sum.
  float p[16];
  float lsum = 0.0f;
  const int base = tid * 16;
#pragma unroll
  for (int j = 0; j < 16; ++j) {
    float a0 = sA[base + j];
    float a1 = sA[L_TOTAL + base + j];
    p[j] = a0 * a0 + a1 * a1;
    lsum += p[j];
  }
  sPart[tid] = lsum;
  __syncthreads();

  // Block scan of the 256 partials.
  float base_excl;
#if defined(__gfx1250__) && __has_builtin(__builtin_amdgcn_wmma_f32_16x16x4_f32)
  // Exact f32 scan via WMMA: P (16x16) times upper-triangular ones, chained
  // as four 16x16x4 f32 WMMAs -> RP[q][r] = sum_{r'<=r} P[q][r'].
  if (tid < 32) {
    const int half = tid >> 4;     // lane group
    const int m    = tid & 15;     // A row (M)
    const int n    = tid & 15;     // C/D column (N)
    v8f acc = {};
#pragma unroll
    for (int c = 0; c < 4; ++c) {
      const int k0 = c * 4 + 2 * half;   // this lane's two K indices: k0, k0+1
      v2f a, bm;
      a[0]  = sPart[m * 16 + k0];
      a[1]  = sPart[m * 16 + k0 + 1];
      bm[0] = (k0     <= n) ? 1.0f : 0.0f;
      bm[1] = (k0 + 1 <= n) ? 1.0f : 0.0f;
      acc = __builtin_amdgcn_wmma_f32_16x16x4_f32(
          /*neg_a=*/false, a, /*neg_b=*/false, bm,
          /*c_mod=*/(short)0, acc, /*reuse_a=*/false, /*reuse_b=*/false);
    }
#pragma unroll
    for (int v = 0; v < 8; ++v) sRP[(v + 8 * half) * 16 + n] = acc[v];
  }
  __syncthreads();
  {
    const int q = tid >> 4, r = tid & 15;
    float off = 0.0f;
    for (int qq = 0; qq < q; ++qq) off += sRP[qq * 16 + 15];
    base_excl = off + ((r > 0) ? sRP[q * 16 + (r - 1)] : 0.0f);
    if (tid == 255) sMisc[0] = off + sRP[15 * 16 + 15];   // grand total
  }
#else
  if (tid == 0) {
    float e = 0.0f;
    for (int t = 0; t < 256; ++t) { sRP[t] = e; e += sPart[t]; }
    sMisc[0] = e;
  }
  __syncthreads();
  base_excl = sRP[tid];
#endif

  // Exclusive prefix S of power.
  {
    float e = 0.0f;
#pragma unroll
    for (int j = 0; j < 16; ++j) { sS[base + j] = base_excl + e; e += p[j]; }
  }
  __syncthreads();

  // Build 64-bit sort keys over sliding-window means (ascending order key).
  unsigned long long* keys = (unsigned long long*)sA;
  for (int i = tid; i < L_TOTAL; i += 256) {
    unsigned long long key;
    if (i < NVALID) {
      float lo = (i >= HALF_WIN) ? sS[i - HALF_WIN] : 0.0f;
      float hi = sS[i + HALF_WIN];               // i+512 <= 3584 < 4096 always
      float sm = (hi - lo) * (1.0f / (float)SEG_LENC);
      uint32_t u = __float_as_uint(sm);
      u ^= (u & 0x80000000u) ? 0xFFFFFFFFu : 0x80000000u;  // order-preserving
      key = ((unsigned long long)u << 32) | (uint32_t)i;
    } else {
      key = 0xFFFFFFFFFFFFFFFFull;               // pad to the top
    }
    keys[i] = key;
  }
  __syncthreads();

  // In-LDS bitonic sort (ascending) of 4096 keys.
  for (int k = 2; k <= L_TOTAL; k <<= 1) {
    for (int j = k >> 1; j > 0; j >>= 1) {
      for (int i = tid; i < L_TOTAL; i += 256) {
        int ixj = i ^ j;
        if (ixj > i) {
          unsigned long long a = keys[i];
          unsigned long long c = keys[ixj];
          bool up = ((i & k) == 0);
          if ((a > c) == up) { keys[i] = c; keys[ixj] = a; }
        }
      }
      __syncthreads();
    }
  }

  // Segment starts: even segments from the high pool (rank-r largest),
  // odd segments from the low pool (rank-r smallest).
  if (tid < NUM_SEGC) {
    const int s = tid;
    const int r = rand_cols[b * NUM_SEGC + s];
    uint32_t idx = (s & 1)
        ? (uint32_t)(keys[r] & 0xFFFFFFFFull)
        : (uint32_t)(keys[(NVALID - 1) - r] & 0xFFFFFFFFull);
    const float a0 = 0.1f  * (float)L_TOTAL;
    const float a1 = 0.78f * (float)L_TOTAL;
    float anchor = a0 + (float)s * ((a1 - a0) / (float)(NUM_SEGC - 1));
    float sf = (1.0f - 0.2f) * anchor + 0.2f * (float)idx
             + ((float)shifts_raw[b * NUM_SEGC + s] - (float)GSHIFT);
    float rf = rintf(sf);                      // round-half-even == jnp.round
    rf = fminf(fmaxf(rf, 0.0f), (float)MAX_START);
    ws_starts[b * NUM_SEGC + s] = (int)rf;
  }
  if (tid == 0) {
    float tot = sMisc[0];                      // sum x^2 over both channels
    ws_amp[b] = 0.08f * sqrtf(tot * (1.0f / (float)(2 * L_TOTAL)) + 1e-12f);
  }
}

// ---------------- Kernel B: streaming writeback with segment additions ------
__global__ __launch_bounds__(256)
void lst_apply_kernel(const float* __restrict__ x,
                      const float* __restrict__ ws_pat,     // 2048 tanh'd
                      const float* __restrict__ ws_scales,  // 4 relu'd
                      const int*   __restrict__ ws_starts,  // [B*4]
                      const float* __restrict__ ws_amp,     // [B]
                      float* __restrict__ out)
{
  __shared__ __align__(16) float spat[2 * SEG_LENC];   // 8 KB
  const int tid = threadIdx.x;
  const int b   = blockIdx.x;

  // Async-stage the tanh'd patterns into LDS (2 * 256 lanes * 16 B = 8 KB).
  {
    uint32_t lds0 = (uint32_t)(uintptr_t)&spat[0];
#pragma unroll
    for (int c = 0; c < 2; ++c) {
      uint32_t off = (uint32_t)(c * 256 + tid) * 16u;
      async_g2l_b128(lds0 + off, ws_pat, off);
    }
    wait_async0();
  }
  __syncthreads();

  const float a = ws_amp[b];
  int   st[NUM_SEGC];
  float cf[NUM_SEGC];
#pragma unroll
  for (int s = 0; s < NUM_SEGC; ++s) {
    st[s] = ws_starts[b * NUM_SEGC + s];
    cf[s] = a * ws_scales[s];
  }

  const float* xr = x   + (size_t)b * (2 * L_TOTAL);
  float*       orow = out + (size_t)b * (2 * L_TOTAL);

  for (int l = tid; l < L_TOTAL; l += 256) {
    float add0 = 0.0f, add1 = 0.0f;
#pragma unroll
    for (int s = 0; s < NUM_SEGC; ++s) {
      int o = l - st[s];
      if ((unsigned)o < (unsigned)SEG_LENC) {
        add0 += cf[s] * spat[o];
        add1 += cf[s] * spat[SEG_LENC + o];
      }
    }
    orow[l]           = xr[l]           + add0;
    orow[L_TOTAL + l] = xr[L_TOTAL + l] + add1;
  }
}

// ---------------- Launch ----------------------------------------------------
extern "C" void kernel_launch(void* const* d_in, const int* in_sizes, int n_in,
                              void* d_out, int out_size, void* d_ws, size_t ws_size,
                              hipStream_t stream) {
  const float* x    = (const float*)d_in[0];   // (4096, 2, 4096)
  const float* pi   = (const float*)d_in[1];   // (1024,)
  const float* pq   = (const float*)d_in[2];   // (1024,)
  const float* ss   = (const float*)d_in[3];   // (4,)
  const int*   rc   = (const int*)  d_in[4];   // (4096, 4)
  const int*   sh   = (const int*)  d_in[5];   // (4096, 4)
  float* out = (float*)d_out;

  const int B = in_sizes[4] / NUM_SEGC;        // 4096 batch rows

  char* ws = (char*)d_ws;
  int*   ws_starts = (int*)  (ws);                                  // B*4 ints
  float* ws_amp    = (float*)(ws + (size_t)B * NUM_SEGC * 4);       // B floats
  float* ws_pat    = (float*)(ws + (size_t)B * NUM_SEGC * 4 + (size_t)B * 4);
  float* ws_scales = (float*)((char*)ws_pat + 2 * SEG_LENC * 4);

  lst_pattern_kernel<<<dim3(8), dim3(256), 0, stream>>>(pi, pq, ss, ws_pat, ws_scales);
  lst_select_kernel<<<dim3(B), dim3(256), 0, stream>>>(x, rc, sh, ws_starts, ws_amp);
  lst_apply_kernel<<<dim3(B), dim3(256), 0, stream>>>(x, ws_pat, ws_scales,
                                                      ws_starts, ws_amp, out);
}